// LogSumExpWirelengthAtomic_33767032881792
// MI455X (gfx1250) — compile-verified
//
#include <hip/hip_runtime.h>
#include <math.h>

#define GAMMA      0.5f
#define INV_GAMMA  2.0f

typedef __attribute__((ext_vector_type(2))) float v2f;
typedef __attribute__((ext_vector_type(8))) float v8f;

// Monotonic float -> uint key mapping (order-preserving), so per-net max can
// use deterministic integer atomics (global_atomic_max_u32 resolved in L2).
__device__ __forceinline__ unsigned f2ord(float f) {
  unsigned u = __float_as_uint(f);
  return (u & 0x80000000u) ? ~u : (u | 0x80000000u);
}
__device__ __forceinline__ float ord2f(unsigned k) {
  unsigned u = (k & 0x80000000u) ? (k & 0x7fffffffu) : ~k;
  return __uint_as_float(u);
}
#define ORD_NEG_INF 0x007FFFFFu   // f2ord(-INFINITY)

// ---------------------------------------------------------------------------
// K0: init workspace (4 max-key arrays to -inf keys, 4 sum arrays to 0) + out
// ---------------------------------------------------------------------------
__global__ void k_init(unsigned* __restrict__ mk, float* __restrict__ sm,
                       long long n4, float* __restrict__ out) {
  long long i = (long long)blockIdx.x * blockDim.x + threadIdx.x;
  if (i < n4) {
    mk[i] = ORD_NEG_INF;
    sm[i] = 0.0f;
  }
  if (i == 0) out[0] = 0.0f;
}

// ---------------------------------------------------------------------------
// K1: per-net max of {x/g, -x/g, y/g, -y/g} via ordered-uint atomicMax.
// Each thread streams 4 contiguous pins with b128 loads.
// ---------------------------------------------------------------------------
__global__ void k_max(const float* __restrict__ pos, const int* __restrict__ map,
                      unsigned* __restrict__ mk, int P, int N) {
  long long t = (long long)blockIdx.x * blockDim.x + threadIdx.x;
  long long i = t * 4;
  if (i + 3 < P) {
    float4 xv = *(const float4*)(pos + i);
    float4 yv = *(const float4*)(pos + (long long)P + i);
    int4   nv = *(const int4*)(map + i);
    float xs[4] = {xv.x, xv.y, xv.z, xv.w};
    float ys[4] = {yv.x, yv.y, yv.z, yv.w};
    int   ns[4] = {nv.x, nv.y, nv.z, nv.w};
#pragma unroll
    for (int j = 0; j < 4; ++j) {
      float a = xs[j] * INV_GAMMA;
      float b = ys[j] * INV_GAMMA;
      int   n = ns[j];
      atomicMax(mk + n,                   f2ord(a));
      atomicMax(mk + (long long)N + n,    f2ord(-a));
      atomicMax(mk + 2LL * N + n,         f2ord(b));
      atomicMax(mk + 3LL * N + n,         f2ord(-b));
    }
  } else if (i < P) {
    for (long long j = i; j < P; ++j) {
      float a = pos[j] * INV_GAMMA;
      float b = pos[(long long)P + j] * INV_GAMMA;
      int   n = map[j];
      atomicMax(mk + n,                   f2ord(a));
      atomicMax(mk + (long long)N + n,    f2ord(-a));
      atomicMax(mk + 2LL * N + n,         f2ord(b));
      atomicMax(mk + 3LL * N + n,         f2ord(-b));
    }
  }
}

// ---------------------------------------------------------------------------
// K2: per-net sum of exp(v - m) via global_atomic_add_f32 (accumulators in L2)
// ---------------------------------------------------------------------------
__global__ void k_expsum(const float* __restrict__ pos, const int* __restrict__ map,
                         const unsigned* __restrict__ mk, float* __restrict__ sm,
                         int P, int N) {
  long long t = (long long)blockIdx.x * blockDim.x + threadIdx.x;
  long long i = t * 4;
  if (i + 3 < P) {
    float4 xv = *(const float4*)(pos + i);
    float4 yv = *(const float4*)(pos + (long long)P + i);
    int4   nv = *(const int4*)(map + i);
    float xs[4] = {xv.x, xv.y, xv.z, xv.w};
    float ys[4] = {yv.x, yv.y, yv.z, yv.w};
    int   ns[4] = {nv.x, nv.y, nv.z, nv.w};
#pragma unroll
    for (int j = 0; j < 4; ++j) {
      float a = xs[j] * INV_GAMMA;
      float b = ys[j] * INV_GAMMA;
      int   n = ns[j];
      float m0 = ord2f(mk[n]);
      float m1 = ord2f(mk[(long long)N + n]);
      float m2 = ord2f(mk[2LL * N + n]);
      float m3 = ord2f(mk[3LL * N + n]);
      atomicAdd(sm + n,                __expf(a - m0));
      atomicAdd(sm + (long long)N + n, __expf(-a - m1));
      atomicAdd(sm + 2LL * N + n,      __expf(b - m2));
      atomicAdd(sm + 3LL * N + n,      __expf(-b - m3));
    }
  } else if (i < P) {
    for (long long j = i; j < P; ++j) {
      float a = pos[j] * INV_GAMMA;
      float b = pos[(long long)P + j] * INV_GAMMA;
      int   n = map[j];
      atomicAdd(sm + n,                __expf(a - ord2f(mk[n])));
      atomicAdd(sm + (long long)N + n, __expf(-a - ord2f(mk[(long long)N + n])));
      atomicAdd(sm + 2LL * N + n,      __expf(b - ord2f(mk[2LL * N + n])));
      atomicAdd(sm + 3LL * N + n,      __expf(-b - ord2f(mk[3LL * N + n])));
    }
  }
}

// ---------------------------------------------------------------------------
// K3: per-net wirelength + global reduction using V_WMMA_F32_16X16X4_F32.
// A = all-ones 16x4, so D[i][j] = C[i][j] + sum_k B[k][j]; every row of D is
// identical, hence sum(all 256 C elems) == 16 * (sum of all values fed via B).
// Each WMMA accumulates 64 net contributions (2 per lane).
// net_contrib is fully branchless (clamped index, unconditional loads, bitwise
// valid mask) so all 9 loads issue together, no exec-mask churn, and EXEC is
// provably all-ones at the WMMA. All indexing is 32-bit (max 4N ~ 16M).
// ---------------------------------------------------------------------------
__device__ __forceinline__ float net_contrib(int i, int N,
                                             const unsigned* __restrict__ mk,
                                             const float* __restrict__ sm,
                                             const unsigned char* __restrict__ mask) {
  int j = (i < N) ? i : (N - 1);           // clamp, no branch
  unsigned char mv = mask[j];              // unconditional load (j in range)
  float acc = 0.0f;
#pragma unroll
  for (int c = 0; c < 4; ++c) {
    float s = sm[c * N + j];
    float m = ord2f(mk[c * N + j]);        // finite whenever s > 0
    float t = m + __logf(s);               // -inf when s == 0 (selected away)
    acc += (s > 0.0f) ? t : 0.0f;
  }
  int valid = (int)(i < N) & (int)(mv != 0);  // bitwise: no short-circuit branch
  return valid ? (GAMMA * acc) : 0.0f;
}

__global__ void k_net_reduce(const unsigned* __restrict__ mk,
                             const float* __restrict__ sm,
                             const unsigned char* __restrict__ mask,
                             float* __restrict__ out, int N) {
  const int lane = threadIdx.x & 31;
  // readfirstlane -> SGPR loop counter: scalar s_cmp loop, EXEC untouched
  // (WMMA requires EXEC all-ones).
  const int wave   = __builtin_amdgcn_readfirstlane(
      (int)((blockIdx.x * blockDim.x + threadIdx.x) >> 5));
  const int nwaves = (int)((gridDim.x * blockDim.x) >> 5);

  v2f A = {1.0f, 1.0f};   // all-ones A matrix (layout-independent reduction)
  v8f C = {};

  const int span   = (int)((((long long)N + 63) / 64) * 64);
  const int stride = nwaves * 64;
  for (int base = wave * 64; base < span; base += stride) {
    int i0 = base + lane * 2;
    v2f B;
    B.x = net_contrib(i0,     N, mk, sm, mask);
    B.y = net_contrib(i0 + 1, N, mk, sm, mask);
    // D = A(ones) * B + C  -> column-sums accumulate into every row of C
    C = __builtin_amdgcn_wmma_f32_16x16x4_f32(
        /*neg_a=*/false, A, /*neg_b=*/false, B,
        /*c_mod=*/(short)0, C, /*reuse_a=*/false, /*reuse_b=*/false);
  }

  float t = C[0] + C[1] + C[2] + C[3] + C[4] + C[5] + C[6] + C[7];
#pragma unroll
  for (int off = 16; off >= 1; off >>= 1) t += __shfl_xor(t, off, 32);
  if (lane == 0) atomicAdd(out, t * 0.0625f);  // /16: rows of D are identical
}

// ---------------------------------------------------------------------------
extern "C" void kernel_launch(void* const* d_in, const int* in_sizes, int n_in,
                              void* d_out, int out_size, void* d_ws, size_t ws_size,
                              hipStream_t stream) {
  const float*         pos  = (const float*)d_in[0];
  const int*           map  = (const int*)d_in[1];
  const unsigned char* mask = (const unsigned char*)d_in[2];
  const int P = in_sizes[1];   // pins
  const int N = in_sizes[2];   // nets

  unsigned* mk = (unsigned*)d_ws;                                          // 4*N keys
  float*    sm = (float*)((char*)d_ws + (size_t)4 * N * sizeof(unsigned)); // 4*N sums
  float*    out = (float*)d_out;

  const long long n4 = 4LL * N;
  const int b_init = (int)((n4 + 255) / 256);
  k_init<<<b_init, 256, 0, stream>>>(mk, sm, n4, out);

  const long long threads = ((long long)P + 3) / 4;
  const int b_pins = (int)((threads + 255) / 256);
  k_max   <<<b_pins, 256, 0, stream>>>(pos, map, mk, P, N);
  k_expsum<<<b_pins, 256, 0, stream>>>(pos, map, mk, sm, P, N);

  k_net_reduce<<<2048, 256, 0, stream>>>(mk, sm, mask, out, N);
}